// QLSTMGen_65481071402165
// MI455X (gfx1250) — compile-verified
//
#include <hip/hip_runtime.h>
#include <hip/hip_bf16.h>
#include <math.h>

// ---------------------------------------------------------------------------
// QLSTM for MI455X (gfx1250, wave32).
//   Sizes: S=1024 (derived), B=128, D=256, N=8 qubits, 4 gates (f,i,u,o).
//   d_in: inputs(S,B,256), Wf(8,264), bf(8), Wi, bi, Wu, bu, Wo, bo
//   d_out: outputs(S,B,8) ++ h_n(128,8) ++ c_n(128,8)
//   d_ws:  [0)      WBP  packed B-panel, K-pair interleaved:
//                     WBP[(k/2)*64 + col*2 + (k&1)] = W_gate[n][k]   (256x32)
//          [8192)   bias 32 f32
//          [16384)  PX   (S*128) x 32 f32  (X-projection + bias, all gates)
// ---------------------------------------------------------------------------

typedef __attribute__((ext_vector_type(2))) float v2f;
typedef __attribute__((ext_vector_type(8))) float v8f;

#define QB 128
#define QD 256
#define QN 8

// ---------------- pack: interleaved WBP + bias32 ----------------------------
__global__ void qlstm_pack(const float* __restrict__ Wf, const float* __restrict__ bf,
                           const float* __restrict__ Wi, const float* __restrict__ bi,
                           const float* __restrict__ Wu, const float* __restrict__ bu,
                           const float* __restrict__ Wo, const float* __restrict__ bo,
                           float* __restrict__ WBP, float* __restrict__ bias32) {
  int t = blockIdx.x * blockDim.x + threadIdx.x;
  if (t < QD * 32) {
    int k = t >> 5, col = t & 31;
    int g = col >> 3, n = col & 7;
    const float* W = (g == 0) ? Wf : (g == 1) ? Wi : (g == 2) ? Wu : Wo;
    // K-pair interleaved so a (k, k+1) fragment pair is one contiguous b64
    WBP[(k >> 1) * 64 + col * 2 + (k & 1)] = W[n * (QD + QN) + k];
  }
  if (t < 32) {
    int g = t >> 3, n = t & 7;
    const float* bb = (g == 0) ? bf : (g == 1) ? bi : (g == 2) ? bu : bo;
    bias32[t] = bb[n];
  }
}

// ---------------- GEMM: PX(M x 32) = X(M x 256) @ W(256 x 32) + bias --------
// V_WMMA_F32_16X16X4_F32, one 16x16 tile per wave, 2 waves/block (N-tiles).
__global__ void qlstm_gemm(const float* __restrict__ X, const float* __restrict__ WBP,
                           const float* __restrict__ bias32, float* __restrict__ PX) {
  const int wave  = threadIdx.x >> 5;     // 0..1 -> N tile
  const int lane  = threadIdx.x & 31;
  const int mtile = blockIdx.x;
  const int l16   = lane & 15;
  const int khalf = (lane >> 4) * 2;      // A/B frag: lanes 16..31 hold K=2,3

  const int row = mtile * 16 + l16;       // A frag row for this lane
  const int col = wave * 16 + l16;        // B frag / output column

  // both bases are 8-byte aligned (khalf even, col*2 even)
  const v2f* __restrict__ xrow = (const v2f*)(X + (size_t)row * QD + khalf);
  const v2f* __restrict__ wcol = (const v2f*)WBP + (size_t)(khalf >> 1) * 32 + col;

  v8f acc = {};
#pragma unroll 8
  for (int k = 0; k < QD; k += 4) {
    v2f a = xrow[k >> 1];                 // X[row][khalf+k .. +1]
    v2f b = wcol[(size_t)(k >> 1) * 32];  // W[khalf+k .. +1][col]
    acc = __builtin_amdgcn_wmma_f32_16x16x4_f32(
        /*neg_a=*/false, a, /*neg_b=*/false, b,
        /*c_mod=*/(short)0, acc, /*reuse_a=*/false, /*reuse_b=*/false);
  }

  // C/D layout: VGPR r, lanes 0-15 -> M=r ; lanes 16-31 -> M=8+r ; N=lane%16
  const int orow = mtile * 16 + (lane >> 4) * 8;
  const float bv = bias32[col];
#pragma unroll
  for (int r = 0; r < 8; ++r)
    PX[(size_t)(orow + r) * 32 + col] = acc[r] + bv;
}

// ---------------- 8-qubit circuit layer (RX, RZ, CNOT) ----------------------
// Amplitude index k = (reg<<5) | lane.  Wire w <-> bit (7-w):
//   w=0,1,2 -> reg bits 2,1,0 (in-register);  w=3..7 -> lane bits 4..0.
template <int w>
__device__ __forceinline__ void qlayer(float (&re)[8], float (&im)[8],
                                       const float (&par)[8], int lane) {
  // ---- RX(par[w]) on wire w:  [[c, -i s],[-i s, c]] ----
  const float th = par[w];
  const float c  = __cosf(0.5f * th);
  const float sn = __sinf(0.5f * th);
  if constexpr (w < 3) {
    constexpr int m = 4 >> w;
#pragma unroll
    for (int r = 0; r < 8; ++r) {
      if (!(r & m)) {
        const int r1 = r | m;
        float a0r = re[r], a0i = im[r], a1r = re[r1], a1i = im[r1];
        re[r]  = c * a0r + sn * a1i;  im[r]  = c * a0i - sn * a1r;
        re[r1] = c * a1r + sn * a0i;  im[r1] = c * a1i - sn * a0r;
      }
    }
  } else {
    constexpr int L = 1 << (7 - w);
#pragma unroll
    for (int r = 0; r < 8; ++r) {
      float pr = __shfl_xor(re[r], L, 32);
      float pi = __shfl_xor(im[r], L, 32);
      re[r] = c * re[r] + sn * pi;
      im[r] = c * im[r] - sn * pr;
    }
  }

  // ---- RZ(par[(w+1)%8]) on wire w: bit0 *= e^{-i t/2}, bit1 *= e^{+i t/2} --
  const float tz = par[(w + 1) & 7];
  const float cz = __cosf(0.5f * tz);
  const float sz = __sinf(0.5f * tz);
  if constexpr (w < 3) {
    constexpr int m = 4 >> w;
#pragma unroll
    for (int r = 0; r < 8; ++r) {
      const float se = (r & m) ? -sz : sz;
      float rr = re[r], ii = im[r];
      re[r] = cz * rr + se * ii;
      im[r] = cz * ii - se * rr;
    }
  } else {
    const float se = ((lane >> (7 - w)) & 1) ? -sz : sz;
#pragma unroll
    for (int r = 0; r < 8; ++r) {
      float rr = re[r], ii = im[r];
      re[r] = cz * rr + se * ii;
      im[r] = cz * ii - se * rr;
    }
  }

  // ---- CNOT(ctrl=w, tgt=(w+1)%8) ----
  if constexpr (w == 0) {            // ctrl reg bit2, tgt reg bit1: swap 4<->6, 5<->7
    float t0;
    t0 = re[4]; re[4] = re[6]; re[6] = t0;  t0 = im[4]; im[4] = im[6]; im[6] = t0;
    t0 = re[5]; re[5] = re[7]; re[7] = t0;  t0 = im[5]; im[5] = im[7]; im[7] = t0;
  } else if constexpr (w == 1) {     // ctrl reg bit1, tgt reg bit0: swap 2<->3, 6<->7
    float t0;
    t0 = re[2]; re[2] = re[3]; re[3] = t0;  t0 = im[2]; im[2] = im[3]; im[3] = t0;
    t0 = re[6]; re[6] = re[7]; re[7] = t0;  t0 = im[6]; im[6] = im[7]; im[7] = t0;
  } else if constexpr (w == 2) {     // ctrl reg bit0, tgt lane bit4: odd regs swap lanes l<->l^16
#pragma unroll
    for (int r = 1; r < 8; r += 2) {
      re[r] = __shfl_xor(re[r], 16, 32);
      im[r] = __shfl_xor(im[r], 16, 32);
    }
  } else if constexpr (w == 7) {     // ctrl lane bit0, tgt reg bit2: odd lanes swap r<->r^4
    const bool odd = lane & 1;
#pragma unroll
    for (int r = 0; r < 4; ++r) {
      float tr = re[r], ti = im[r];
      re[r]     = odd ? re[r + 4] : re[r];
      im[r]     = odd ? im[r + 4] : im[r];
      re[r + 4] = odd ? tr : re[r + 4];
      im[r + 4] = odd ? ti : im[r + 4];
    }
  } else {                           // w=3..6: ctrl lane bit (7-w), tgt lane bit (6-w)
    constexpr int Lt = 1 << (6 - w);
    const bool ctl = (lane >> (7 - w)) & 1;
#pragma unroll
    for (int r = 0; r < 8; ++r) {
      float tr = __shfl_xor(re[r], Lt, 32);
      float ti = __shfl_xor(im[r], Lt, 32);
      re[r] = ctl ? tr : re[r];
      im[r] = ctl ? ti : im[r];
    }
  }
}

// ---------------- scan: 1 block per batch element, 1 wave per gate ----------
__global__ void qlstm_scan(const float* __restrict__ PX,
                           const float* __restrict__ Wf, const float* __restrict__ Wi,
                           const float* __restrict__ Wu, const float* __restrict__ Wo,
                           float* __restrict__ out, int S) {
  const int b    = blockIdx.x;          // 0..127
  const int g    = threadIdx.x >> 5;    // 0..3  (f,i,u,o)
  const int lane = threadIdx.x & 31;
  const float* __restrict__ Wg = (g == 0) ? Wf : (g == 1) ? Wi : (g == 2) ? Wu : Wo;

  __shared__ float shh[QN];
  __shared__ float shc[QN];
  __shared__ float shg[4 * QN];

  // hoist recurrent weights off the serial path: lane n holds Wh_g[n][0..7]
  float wh[QN];
#pragma unroll
  for (int j = 0; j < QN; ++j)
    wh[j] = (lane < QN) ? Wg[lane * (QD + QN) + QD + j] : 0.f;

  if (threadIdx.x < QN) { shh[threadIdx.x] = 0.f; shc[threadIdx.x] = 0.f; }
  __syncthreads();

  for (int s = 0; s < S; ++s) {
    // keep next step's PX cacheline warm (global_prefetch_b8)
    if (lane == 0 && s + 1 < S)
      __builtin_prefetch(&PX[((size_t)(s + 1) * QB + b) * 32], 0, 0);

    // params[n] = PX[s,b,g*8+n] (incl. bias) + h @ Wh_g[n,:]
    float par = 0.f;
    if (lane < QN) {
      par = PX[((size_t)s * QB + b) * 32 + g * QN + lane];
#pragma unroll
      for (int j = 0; j < QN; ++j)
        par += shh[j] * wh[j];
    }
    float pars[8];
#pragma unroll
    for (int k = 0; k < 8; ++k) pars[k] = __shfl(par, k, 32);

    // statevector: 256 complex amps, 8 per lane
    float re[8], im[8];
#pragma unroll
    for (int r = 0; r < 8; ++r) { re[r] = 0.f; im[r] = 0.f; }
    re[0] = (lane == 0) ? 1.f : 0.f;

    qlayer<0>(re, im, pars, lane);
    qlayer<1>(re, im, pars, lane);
    qlayer<2>(re, im, pars, lane);
    qlayer<3>(re, im, pars, lane);
    qlayer<4>(re, im, pars, lane);
    qlayer<5>(re, im, pars, lane);
    qlayer<6>(re, im, pars, lane);
    qlayer<7>(re, im, pars, lane);

    // probs of basis states 0..7 live in reg 0, lanes 0..7
    float p = re[0] * re[0] + im[0] * im[0];
    float sum = p;
    sum += __shfl_xor(sum, 1, 32);
    sum += __shfl_xor(sum, 2, 32);
    sum += __shfl_xor(sum, 4, 32);
    float prob = p / sum;
    if (g == 2) prob = tanhf(prob);     // u-gate has tanh applied
    if (lane < QN) shg[g * QN + lane] = prob;
    __syncthreads();

    if (threadIdx.x < QN) {
      const int n = threadIdx.x;
      float fv = shg[n], iv = shg[QN + n], gv = shg[2 * QN + n], ov = shg[3 * QN + n];
      float cn = fv * shc[n] + iv * gv;
      float hn = ov * tanhf(cn);
      shc[n] = cn;
      shh[n] = hn;
      out[((size_t)s * QB + b) * QN + n] = hn;
    }
    __syncthreads();
  }

  if (threadIdx.x < QN) {
    const int n = threadIdx.x;
    const size_t base = (size_t)S * QB * QN;
    out[base + (size_t)b * QN + n] = shh[n];                   // h_n
    out[base + (size_t)QB * QN + (size_t)b * QN + n] = shc[n]; // c_n
  }
}

// ---------------------------------------------------------------------------
extern "C" void kernel_launch(void* const* d_in, const int* in_sizes, int n_in,
                              void* d_out, int out_size, void* d_ws, size_t ws_size,
                              hipStream_t stream) {
  (void)n_in; (void)out_size; (void)ws_size;
  const float* X  = (const float*)d_in[0];
  const float* Wf = (const float*)d_in[1];
  const float* bf = (const float*)d_in[2];
  const float* Wi = (const float*)d_in[3];
  const float* bi = (const float*)d_in[4];
  const float* Wu = (const float*)d_in[5];
  const float* bu = (const float*)d_in[6];
  const float* Wo = (const float*)d_in[7];
  const float* bo = (const float*)d_in[8];

  const int S = in_sizes[0] / (QB * QD);

  float* ws     = (float*)d_ws;
  float* WBP    = ws;             // 256*32 floats, K-pair interleaved
  float* bias32 = ws + 8192;      // 32 floats
  float* PX     = ws + 16384;     // S*128*32 floats

  qlstm_pack<<<32, 256, 0, stream>>>(Wf, bf, Wi, bi, Wu, bu, Wo, bo, WBP, bias32);

  const int mtiles = (S * QB) / 16;           // 8192 for S=1024
  qlstm_gemm<<<mtiles, 64, 0, stream>>>(X, WBP, bias32, PX);

  qlstm_scan<<<QB, 128, 0, stream>>>(PX, Wf, Wi, Wu, Wo, (float*)d_out, S);
}